// STDP_CA3_17635135717851
// MI455X (gfx1250) — compile-verified
//
#include <hip/hip_runtime.h>

// ---------------- constants from the reference ----------------
#define NSTATES 4096
#define GAMMA   0.5f    // GAMMA_M0
#define GT      0.99f   // GAMMA_T
#define BOFF    0.99f   // B_OFFSET
#define NITER   32      // Neumann iterations: 0.5^32 << fp32 eps

typedef __attribute__((ext_vector_type(2))) float v2f;
typedef __attribute__((ext_vector_type(8))) float v8f;

// ---------------- O(N) prep: a0=input, eta_invs_n, real_T_count_n ----------------
__global__ __launch_bounds__(256) void prep_kernel(
    const float* __restrict__ input, const float* __restrict__ eta_invs,
    const float* __restrict__ tcount, const int* __restrict__ prev,
    float* __restrict__ a0, float* __restrict__ out_eta, float* __restrict__ out_cnt) {
  int i = blockIdx.x * blockDim.x + threadIdx.x;
  if (i >= NSTATES) return;
  a0[i] = input[i];
  float pb = prev[i] > 0 ? 1.f : 0.f;
  out_eta[i] = (float)prev[i] + GT * eta_invs[i];
  out_cnt[i] = GT * tcount[i] + pb;
}

// ---------------- y = rhs + GAMMA * (J @ x) via fp32 WMMA ----------------
// One block (8 waves) per 16-row strip of J. Each wave covers a 512-wide K
// chunk with V_WMMA_F32_16X16X4_F32: A = 16x4 J tile, B column 0 = x chunk,
// C column 0 (lanes 0 and 16) accumulates exact fp32 row dot-products.
// x is staged into LDS with GLOBAL_LOAD_ASYNC_TO_LDS_B128 (ASYNCcnt path).
__global__ __launch_bounds__(256) void matvec_wmma(
    const float* __restrict__ J, const float* __restrict__ x,
    const float* __restrict__ rhs, float* __restrict__ y) {
  __shared__ float xs[NSTATES];   // 16 KB: x staged once, reused by all 8 waves
  __shared__ float red[128];      // 8 waves * 16 row-partials
  const int tid  = threadIdx.x;
  const int wave = tid >> 5;
  const int lane = tid & 31;
  const int strip = blockIdx.x;           // 0..255

  // Async DMA x -> LDS (b128 per lane, 4 rounds of 256 threads = 16 KB)
  for (int i = tid; i < NSTATES / 4; i += 256) {
    unsigned ldsoff = (unsigned)(size_t)(xs + 4 * i);     // low 32 bits = LDS byte addr
    const float4* gsrc = ((const float4*)x) + i;
    asm volatile("global_load_async_to_lds_b128 %0, %1, off"
                 :: "v"(ldsoff), "v"(gsrc) : "memory");
  }
  asm volatile("s_wait_asynccnt 0x0" ::: "memory");
  __syncthreads();

  // A-matrix 16x4 fp32 layout: lanes 0-15 hold M=0..15 K={0,1}; lanes 16-31 K={2,3}
  const int row  = (strip << 4) + (lane & 15);
  const float* Jrow = J + row * NSTATES;
  const int koff = (lane >> 4) << 1;          // 0 or 2
  const float sel = ((lane & 15) == 0) ? 1.f : 0.f;  // B column 0 only

  v8f c0 = {}; v8f c1 = {};
  const int kbeg = wave << 9;                 // 512-wide K chunk per wave
  for (int k = kbeg; k < kbeg + 512; k += 8) {
    v2f a0 = *(const v2f*)(Jrow + k + koff);
    v2f x0 = *(const v2f*)(xs   + k + koff);
    v2f b0; b0.x = sel * x0.x; b0.y = sel * x0.y;
    c0 = __builtin_amdgcn_wmma_f32_16x16x4_f32(false, a0, false, b0,
                                               (short)0, c0, false, false);
    v2f a1 = *(const v2f*)(Jrow + k + 4 + koff);
    v2f x1 = *(const v2f*)(xs   + k + 4 + koff);
    v2f b1; b1.x = sel * x1.x; b1.y = sel * x1.y;
    c1 = __builtin_amdgcn_wmma_f32_16x16x4_f32(false, a1, false, b1,
                                               (short)0, c1, false, false);
  }
  v8f c = c0 + c1;

  // C column N=0: lane0 VGPR r -> row strip*16+r (r=0..7); lane16 -> rows +8..15
  if ((lane & 15) == 0) {
    int base = (wave << 4) + ((lane >> 4) << 3);
#pragma unroll
    for (int r = 0; r < 8; ++r) red[base + r] = c[r];
  }
  __syncthreads();

  if (tid < 16) {
    float s = 0.f;
#pragma unroll
    for (int w = 0; w < 8; ++w) s += red[(w << 4) + tid];
    int orow = (strip << 4) + tid;
    y[orow] = rhs[orow] + GAMMA * s;
  }
}

// ---------------- O(N) traces from activity ----------------
__global__ __launch_bounds__(256) void post_kernel(
    const float* __restrict__ act, const float* __restrict__ Bpos_in,
    const float* __restrict__ Bneg_in, float* __restrict__ Bpos_out,
    float* __restrict__ Bneg_out) {
  int i = blockIdx.x * blockDim.x + threadIdx.x;
  if (i >= NSTATES) return;
  float X  = fminf(fmaxf(act[i], 0.f), 1.f);
  float a01 = (X >= BOFF) ? 1.f : 0.f;
  const float lrp = 0.1f / 0.12f;               // DT/TAU_POS
  Bpos_out[i] = fminf((1.f - lrp) * Bpos_in[i] + lrp * 7.f * a01, 6.f);
  const float lrn = 0.1f;                       // DT/TAU_NEG, A_NEG = 0
  Bneg_out[i] = (1.f - lrn) * Bneg_in[i] + lrn * 0.f * a01;
}

// ---------------- O(N^2) J update (float4 / b128, HBM-bound) ----------------
__global__ __launch_bounds__(256) void j_update_kernel(
    const float* __restrict__ Jin, float* __restrict__ Jout,
    const float* __restrict__ act, const float* __restrict__ Bpos,
    const float* __restrict__ Bneg, const float* __restrict__ etainv,
    const int* __restrict__ prev) {
  const int nv = NSTATES * (NSTATES / 4);
  const int stride = gridDim.x * blockDim.x;
  for (int idx = blockIdx.x * blockDim.x + threadIdx.x; idx < nv; idx += stride) {
    if (idx + stride < nv)                       // cover HBM latency for next chunk
      __builtin_prefetch(((const float4*)Jin) + idx + stride, 0, 0);
    const int i  = idx >> 10;             // row
    const int j4 = (idx & 1023) << 2;     // first col of this float4
    float4 Jv = ((const float4*)Jin)[idx];
    // per-row factors (cache-resident small vectors)
    float Xi    = fminf(fmaxf(act[i], 0.f), 1.f);
    float actXi = (Xi < BOFF) ? 0.f : Xi;
    float Bni   = fmaxf(Bneg[i], 0.f);
    // per-column factors
    float4 a4  = ((const float4*)act)[idx & 1023];
    float4 bp4 = ((const float4*)Bpos)[idx & 1023];
    float4 ei4 = ((const float4*)etainv)[idx & 1023];
    int4   pv4 = ((const int4*)prev)[idx & 1023];
    float Jc[4] = {Jv.x, Jv.y, Jv.z, Jv.w};
    float ac[4] = {a4.x, a4.y, a4.z, a4.w};
    float bp[4] = {bp4.x, bp4.y, bp4.z, bp4.w};
    float ei[4] = {ei4.x, ei4.y, ei4.z, ei4.w};
    int   pv[4] = {pv4.x, pv4.y, pv4.z, pv4.w};
    float outv[4];
#pragma unroll
    for (int c = 0; c < 4; ++c) {
      int j = j4 + c;
      float Xj    = fminf(fmaxf(ac[c], 0.f), 1.f);
      float actXj = (Xj < BOFF) ? 0.f : Xj;
      float Bpj   = fmaxf(bp[c], 0.f);
      float eta   = 1.f / ei[c];
      float upd = (j == i) ? (0.1f * actXi * Bpj * 1.65f)
                           : (0.1f * (actXi * Bpj + actXj * Bni) * 10.08f);
      float cand = fminf(fmaxf((1.f - eta) * Jc[c] + eta * upd, 0.f), 1.f);
      outv[c] = (pv[c] == 1) ? cand : Jc[c];
    }
    ((float4*)Jout)[idx] = make_float4(outv[0], outv[1], outv[2], outv[3]);
  }
}

// ---------------- O(N^2) real_T_tilde update ----------------
__global__ __launch_bounds__(256) void t_update_kernel(
    const float* __restrict__ Tin, float* __restrict__ Tout,
    const int* __restrict__ prev, const int* __restrict__ curr) {
  const int nv = NSTATES * (NSTATES / 4);
  const int stride = gridDim.x * blockDim.x;
  for (int idx = blockIdx.x * blockDim.x + threadIdx.x; idx < nv; idx += stride) {
    if (idx + stride < nv)
      __builtin_prefetch(((const float4*)Tin) + idx + stride, 0, 0);
    const int i = idx >> 10;
    float pb = prev[i] > 0 ? 1.f : 0.f;
    float4 t4 = ((const float4*)Tin)[idx];
    int4  c4  = ((const int4*)curr)[idx & 1023];
    float4 o;
    o.x = GT * t4.x + pb * (c4.x > 0 ? 1.f : 0.f);
    o.y = GT * t4.y + pb * (c4.y > 0 ? 1.f : 0.f);
    o.z = GT * t4.z + pb * (c4.z > 0 ? 1.f : 0.f);
    o.w = GT * t4.w + pb * (c4.w > 0 ? 1.f : 0.f);
    ((float4*)Tout)[idx] = o;
  }
}

// ---------------- host launcher ----------------
extern "C" void kernel_launch(void* const* d_in, const int* in_sizes, int n_in,
                              void* d_out, int out_size, void* d_ws, size_t ws_size,
                              hipStream_t stream) {
  const int N = NSTATES;
  const float* input = (const float*)d_in[0];
  const float* J     = (const float*)d_in[1];
  const float* Bpos  = (const float*)d_in[2];
  const float* Bneg  = (const float*)d_in[3];
  const float* etai  = (const float*)d_in[4];
  const float* Ttil  = (const float*)d_in[5];
  const float* Tcnt  = (const float*)d_in[6];
  const int*   prev  = (const int*)d_in[7];
  const int*   curr  = (const int*)d_in[8];

  // output layout: activity, J_n, B_pos, B_neg, eta_invs, T_tilde, T_count
  float* out    = (float*)d_out;
  float* o_act  = out;
  float* o_J    = out + N;
  float* o_Bpos = o_J + (size_t)N * N;
  float* o_Bneg = o_Bpos + N;
  float* o_eta  = o_Bneg + N;
  float* o_Ttil = o_eta + N;
  float* o_Tcnt = o_Ttil + (size_t)N * N;

  float* a0 = (float*)d_ws;       // ping-pong iterates: 2*N floats = 32 KB
  float* a1 = a0 + N;

  prep_kernel<<<N / 256, 256, 0, stream>>>(input, etai, Tcnt, prev, a0, o_eta, o_Tcnt);

  // a <- input + GAMMA * J * a  (Neumann series for (I - GAMMA*J)^{-1} input)
  float* xv = a0;
  for (int it = 0; it < NITER; ++it) {
    float* yv = (it == NITER - 1) ? o_act : ((xv == a0) ? a1 : a0);
    matvec_wmma<<<N / 16, 256, 0, stream>>>(J, xv, input, yv);
    xv = yv;
  }

  post_kernel<<<N / 256, 256, 0, stream>>>(o_act, Bpos, Bneg, o_Bpos, o_Bneg);
  j_update_kernel<<<8192, 256, 0, stream>>>(J, o_J, o_act, o_Bpos, o_Bneg, o_eta, prev);
  t_update_kernel<<<8192, 256, 0, stream>>>(Ttil, o_Ttil, prev, curr);
}